// InteractionSitesModule_16578573763158
// MI455X (gfx1250) — compile-verified
//
#include <hip/hip_runtime.h>
#include <hip/hip_bf16.h>

typedef __attribute__((ext_vector_type(2))) float v2f;
typedef __attribute__((ext_vector_type(8))) float v8f;

#define HDIM 1024
#define DDIM 64
#define KC   64
#define WSTR 68   // LDS stride: bank = (4n + k) % 64 -> halves on disjoint banks
#define ASTR 68

__device__ __forceinline__ v8f wmma_f32_k4(v2f a, v2f b, v8f c) {
    return __builtin_amdgcn_wmma_f32_16x16x4_f32(
        /*neg_a=*/false, a, /*neg_b=*/false, b,
        /*c_mod=*/(short)0, c, /*reuse_a=*/false, /*reuse_b=*/false);
}

__device__ __forceinline__ float gelu_exact(float x) {
    return 0.5f * x * (1.0f + erff(x * 0.70710678118654752f));
}

// ---------------------------------------------------------------------------
// Fused dual projection: for ONE input X compute BOTH
//   out0 = gelu(X@W0 + b0) * mask ,  out1 = gelu(X@W1 + b1) * mask
// Halves the dominant HBM read (X) vs. two separate launches.
// block = 256 threads (8 waves); block covers 128 rows x 64 cols.
// rows multiple of 128 (4096, 32000 both are).
// W tiles stored TRANSPOSED in LDS so each B fragment is one aligned b64 load.
// ---------------------------------------------------------------------------
__global__ void __launch_bounds__(256)
proj2_kernel(const float* __restrict__ X,
             const float* __restrict__ W0, const float* __restrict__ bias0,
             const float* __restrict__ W1, const float* __restrict__ bias1,
             const float* __restrict__ mask,
             float* __restrict__ out0, float* __restrict__ out1)
{
    __shared__ float shA [128 * ASTR];
    __shared__ float shW0[DDIM * WSTR];   // transposed: [n][k]
    __shared__ float shW1[DDIM * WSTR];

    const int tid  = threadIdx.x;
    const int wave = tid >> 5;
    const int lane = tid & 31;
    const int half = lane >> 4;
    const int lm   = lane & 15;
    const int row0 = blockIdx.x * 128;

    v8f acc0[4], acc1[4];
    #pragma unroll
    for (int nt = 0; nt < 4; ++nt)
        #pragma unroll
        for (int v = 0; v < 8; ++v) { acc0[nt][v] = 0.0f; acc1[nt][v] = 0.0f; }

    for (int k0 = 0; k0 < HDIM; k0 += KC) {
        __syncthreads();
        // W chunks (KC x 64 in memory) -> transposed [n][k] in LDS, coalesced reads
        for (int idx = tid; idx < KC * DDIM; idx += 256) {
            const int kk = idx >> 6, n = idx & 63;
            shW0[n * WSTR + kk] = W0[(size_t)(k0 + kk) * DDIM + n];
            shW1[n * WSTR + kk] = W1[(size_t)(k0 + kk) * DDIM + n];
        }
        // A chunk (128 rows x KC), coalesced
        for (int idx = tid; idx < 128 * KC; idx += 256) {
            const int r = idx >> 6, kk = idx & 63;
            shA[r * ASTR + kk] = X[(size_t)(row0 + r) * HDIM + k0 + kk];
        }
        __syncthreads();

        const float* aBase  = &shA [(wave * 16 + lm) * ASTR + 2 * half];
        const float* wBase0 = &shW0[lm * WSTR + 2 * half];
        const float* wBase1 = &shW1[lm * WSTR + 2 * half];
        #pragma unroll 4
        for (int kk = 0; kk < KC; kk += 4) {
            // A frag: lane M=lm, VGPR v <-> K = kk + v + 2*half  (one b64 load)
            const v2f a = *(const v2f*)(aBase + kk);
            #pragma unroll
            for (int nt = 0; nt < 4; ++nt) {
                // B frag: lane N=nt*16+lm, VGPR v <-> K = kk + v + 2*half (one b64 load)
                const v2f bf0 = *(const v2f*)(wBase0 + nt * 16 * WSTR + kk);
                const v2f bf1 = *(const v2f*)(wBase1 + nt * 16 * WSTR + kk);
                acc0[nt] = wmma_f32_k4(a, bf0, acc0[nt]);
                acc1[nt] = wmma_f32_k4(a, bf1, acc1[nt]);
            }
        }
    }

    // Epilogue: bias + exact GELU + row mask.  C layout: VGPR v -> row v+8*half.
    const int rbase = row0 + wave * 16;
    #pragma unroll
    for (int nt = 0; nt < 4; ++nt) {
        const int n   = nt * 16 + lm;
        const float bv0 = bias0[n];
        const float bv1 = bias1[n];
        #pragma unroll
        for (int v = 0; v < 8; ++v) {
            const int grow = rbase + v + 8 * half;
            const float mk = mask[grow];
            const float x0 = acc0[nt][v] + bv0;
            const float x1 = acc1[nt][v] + bv1;
            out0[(size_t)grow * DDIM + n] = gelu_exact(x0) * mk;
            out1[(size_t)grow * DDIM + n] = gelu_exact(x1) * mk;
        }
    }
}

// ---------------------------------------------------------------------------
// scores = pc @ pp^T (per batch), maps = sigmoid(scores)*cm*pm,
// masked = scores*pairwise_masks -> row sums (-> vec_c partial via fc)
//                                 -> col sums (-> vec_p partial via fp)
// grid = 256 blocks = (b, itile); block = 256 threads = 8 waves over j-tiles.
// ---------------------------------------------------------------------------
__global__ void __launch_bounds__(256)
score_kernel(const float* __restrict__ pc, const float* __restrict__ pp,
             const float* __restrict__ fc, const float* __restrict__ fp,
             const float* __restrict__ pwmask, const float* __restrict__ cmask,
             const float* __restrict__ pmask, float* __restrict__ maps,
             float* __restrict__ ws_c, float* __restrict__ ws_p)
{
    const int b    = blockIdx.x >> 4;
    const int it   = blockIdx.x & 15;
    const int tid  = threadIdx.x;
    const int wave = tid >> 5;
    const int lane = tid & 31;
    const int half = lane >> 4;
    const int lm   = lane & 15;

    __shared__ float rs_sh[8][16];
    __shared__ float vp_sh[8][64];
    __shared__ float rs_tot[16];

    // Preload A fragments for this i-tile: 16 K-steps of pc[i, :64]
    v2f afr[16];
    const float* pcb = pc + ((size_t)(b * 256 + it * 16 + lm)) * DDIM + 2 * half;
    #pragma unroll
    for (int s = 0; s < 16; ++s)
        afr[s] = *(const v2f*)(pcb + s * 4);

    float rs[8];
    #pragma unroll
    for (int v = 0; v < 8; ++v) rs[v] = 0.0f;
    float vp0 = 0.0f, vp1 = 0.0f;

    for (int jt = wave; jt < 125; jt += 8) {
        const int j0 = jt * 16;
        v8f c;
        #pragma unroll
        for (int v = 0; v < 8; ++v) c[v] = 0.0f;

        // B frag from pp^T: lane N(col)=lm, VGPR v <-> K = s*4 + v + 2*half
        const float* ppb = pp + ((size_t)(b * 2000 + j0 + lm)) * DDIM + 2 * half;
        #pragma unroll
        for (int s = 0; s < 16; ++s) {
            const v2f bf = *(const v2f*)(ppb + s * 4);
            c = wmma_f32_k4(afr[s], bf, c);
        }

        const float pmv = pmask[b * 2000 + j0 + lm];
        float colp = 0.0f;
        #pragma unroll
        for (int v = 0; v < 8; ++v) {
            const int i = it * 16 + v + 8 * half;
            const size_t idx = ((size_t)b * 256 + i) * 2000 + j0 + lm;
            const float s   = c[v];
            const float sig = 1.0f / (1.0f + expf(-s));
            // streaming 33MB write / 33MB read: non-temporal, keep L2 for pp/fp
            __builtin_nontemporal_store(sig * cmask[b * 256 + i] * pmv, &maps[idx]);
            const float m = s * __builtin_nontemporal_load(&pwmask[idx]);
            rs[v] += m;
            colp  += m;
        }
        // complete column sums (cross-half add); both halves end identical
        colp += __shfl_xor(colp, 16, 32);

        // vec_p partial: vp[d] += sum_col cs[col] * fp[j0+col, d]; d = lane, lane+32
        #pragma unroll
        for (int col = 0; col < 16; ++col) {
            const float csv = __shfl(colp, col, 32);
            const float* fpr = fp + ((size_t)(b * 2000 + j0 + col)) * DDIM;
            vp0 += csv * fpr[lane];
            vp1 += csv * fpr[lane + 32];
        }
    }

    // Reduce row sums across 16 columns (lanes) within each half
    #pragma unroll
    for (int v = 0; v < 8; ++v) {
        float r = rs[v];
        r += __shfl_xor(r, 8, 16);
        r += __shfl_xor(r, 4, 16);
        r += __shfl_xor(r, 2, 16);
        r += __shfl_xor(r, 1, 16);
        rs[v] = r;
    }
    if (lm == 0) {
        #pragma unroll
        for (int v = 0; v < 8; ++v) rs_sh[wave][v + 8 * half] = rs[v];
    }
    vp_sh[wave][lane]      = vp0;
    vp_sh[wave][lane + 32] = vp1;
    __syncthreads();

    if (tid < 16) {  // total row sums for this (b, itile), fixed order
        float s = 0.0f;
        #pragma unroll
        for (int w = 0; w < 8; ++w) s += rs_sh[w][tid];
        rs_tot[tid] = s;
    }
    if (tid < 64) {  // vec_p partial for this block, fixed order
        float s = 0.0f;
        #pragma unroll
        for (int w = 0; w < 8; ++w) s += vp_sh[w][tid];
        ws_p[(size_t)blockIdx.x * 64 + tid] = s;
    }
    __syncthreads();

    if (tid < 64) {  // vec_c partial = sum_r rs_tot[r] * fc[i0+r, d]
        float s = 0.0f;
        const float* fcb = fc + ((size_t)(b * 256 + it * 16)) * DDIM + tid;
        #pragma unroll
        for (int r = 0; r < 16; ++r) s += rs_tot[r] * fcb[r * DDIM];
        ws_c[(size_t)blockIdx.x * 64 + tid] = s;
    }
}

// Sum the 16 per-i-tile partials into the concatenated (16,128) vector output.
__global__ void __launch_bounds__(128)
vec_reduce(const float* __restrict__ ws_c, const float* __restrict__ ws_p,
           float* __restrict__ vec_out)
{
    const int b = blockIdx.x;
    const int d = threadIdx.x;   // 0..127
    float s = 0.0f;
    if (d < 64) {
        #pragma unroll
        for (int it = 0; it < 16; ++it) s += ws_c[((size_t)(b * 16 + it)) * 64 + d];
    } else {
        const int dd = d - 64;
        #pragma unroll
        for (int it = 0; it < 16; ++it) s += ws_p[((size_t)(b * 16 + it)) * 64 + dd];
    }
    vec_out[b * 128 + d] = s;
}

extern "C" void kernel_launch(void* const* d_in, const int* in_sizes, int n_in,
                              void* d_out, int out_size, void* d_ws, size_t ws_size,
                              hipStream_t stream) {
    (void)in_sizes; (void)n_in; (void)out_size; (void)ws_size;
    const float* prot   = (const float*)d_in[0];   // (16, 2000, 1024)
    const float* pmask  = (const float*)d_in[1];   // (16, 2000)
    const float* comp   = (const float*)d_in[2];   // (16, 256, 1024)
    const float* cmask  = (const float*)d_in[3];   // (16, 256)
    const float* pwmask = (const float*)d_in[4];   // (16, 256, 2000)
    const float* W_pc = (const float*)d_in[5];
    const float* b_pc = (const float*)d_in[6];
    const float* W_pp = (const float*)d_in[7];
    const float* b_pp = (const float*)d_in[8];
    const float* W_fc = (const float*)d_in[9];
    const float* b_fc = (const float*)d_in[10];
    const float* W_fp = (const float*)d_in[11];
    const float* b_fp = (const float*)d_in[12];

    float* out  = (float*)d_out;
    float* vec  = out;                       // (16, 128)
    float* maps = out + 2048;                // (16, 256, 2000)
    float* pc   = out + 2048 + 8192000;      // (16, 256, 64)
    float* ppo  = pc + 262144;               // (16, 2000, 64)
    float* fc   = ppo + 2048000;             // (16, 256, 64)
    float* fpo  = fc + 262144;               // (16, 2000, 64)

    float* ws_c = (float*)d_ws;              // 16*16*64 floats
    float* ws_p = ws_c + 16 * 16 * 64;       // 16*16*64 floats

    // Fused projections: each input read from HBM exactly once
    proj2_kernel<<<32,  256, 0, stream>>>(comp, W_pc, b_pc, W_fc, b_fc, cmask, pc,  fc);
    proj2_kernel<<<250, 256, 0, stream>>>(prot, W_pp, b_pp, W_fp, b_fp, pmask, ppo, fpo);

    // Scores, maps, masked row/col sums -> per-block partial vectors
    score_kernel<<<256, 256, 0, stream>>>(pc, ppo, fc, fpo, pwmask, cmask, pmask,
                                          maps, ws_c, ws_p);

    // Final (16,128) concat [vec_c | vec_p]
    vec_reduce<<<16, 128, 0, stream>>>(ws_c, ws_p, vec);
}